// pd_mask_12283606468117
// MI455X (gfx1250) — compile-verified
//
#include <hip/hip_runtime.h>

// out[b,t,d] = x[b,t,d] * (t % PERIOD != mask_idx)
// B=8, T=8192, D=512, PERIOD=4, f32.
//
// AI ~ 0 FLOP/byte -> HBM-bound (23.3 TB/s). Traffic floor with masked-row
// read skipping: 96 MiB read + 128 MiB write = 224 MiB ~ 9.6 us.
//
// Mapping: block = 8 rows (2 KiB each); threads 0-127 cover rows 0..3 of the
// tile, threads 128-255 rows 4..7; thread's k-th chunk = same column of row k,
// so tmod == k at compile time. We dispatch ONCE on the (scalar, wave-uniform)
// mask value m into a specialized straight-line body with the masked row as a
// template constant: one basic block = loads can clause, stores pipeline
// (loadcnt<=2,1,0), and there are no CFG-merge phi copies (the v_mov_b64
// storm the previous revision suffered).

typedef float __attribute__((ext_vector_type(4))) f4;

static constexpr int PERIOD         = 4;
static constexpr int D              = 512;
static constexpr int CHUNKS_PER_ROW = D / 4;   // 128 float4 chunks per 2KiB row
static constexpr int THREADS        = 256;

// M = masked row (compile-time). M == -1 => keep everything.
template <int M>
__device__ __forceinline__ void do_tile(const f4* __restrict__ x,
                                        f4* __restrict__ out,
                                        unsigned base)
{
    f4 v[PERIOD];
#pragma unroll
    for (int k = 0; k < PERIOD; ++k) {
        if (k == M)
            v[k] = (f4){0.f, 0.f, 0.f, 0.f};
        else
            v[k] = x[base + (unsigned)k * CHUNKS_PER_ROW];  // RT: L2-resident reads
    }
#pragma unroll
    for (int k = 0; k < PERIOD; ++k)
        __builtin_nontemporal_store(v[k], &out[base + (unsigned)k * CHUNKS_PER_ROW]);
}

__global__ __launch_bounds__(THREADS) void pd_mask_kernel(
    const f4* __restrict__ x,
    const int* __restrict__ mask_idx_p,
    f4* __restrict__ out)
{
    // Uniform scalar in SGPR (s_load_b32).
    const int m = __builtin_amdgcn_readfirstlane(*mask_idx_p);

    const unsigned tid  = threadIdx.x;
    const unsigned half = tid >> 7;                    // which 4-row group
    const unsigned col  = tid & (CHUNKS_PER_ROW - 1);  // column within row
    const unsigned base = blockIdx.x * (THREADS * PERIOD)
                        + half * (PERIOD * CHUNKS_PER_ROW)
                        + col;

    switch (m) {
        case 0:  do_tile<0>(x, out, base); break;
        case 1:  do_tile<1>(x, out, base); break;
        case 2:  do_tile<2>(x, out, base); break;
        case 3:  do_tile<3>(x, out, base); break;
        default: do_tile<-1>(x, out, base); break;  // out-of-range: keep all
    }
}

extern "C" void kernel_launch(void* const* d_in, const int* in_sizes, int n_in,
                              void* d_out, int out_size, void* d_ws, size_t ws_size,
                              hipStream_t stream) {
    const f4*  x        = (const f4*)d_in[0];
    const int* mask_idx = (const int*)d_in[1];
    f4*        out      = (f4*)d_out;

    const unsigned total_f32 = (unsigned)in_sizes[0];          // 33,554,432
    const unsigned n_chunks  = total_f32 / 4;                  // 8,388,608
    const unsigned blocks    = n_chunks / (THREADS * PERIOD);  // 8192, exact

    pd_mask_kernel<<<blocks, THREADS, 0, stream>>>(x, mask_idx, out);
}